// SST_GCN_block_25967372272308
// MI455X (gfx1250) — compile-verified
//
#include <hip/hip_runtime.h>
#include <hip/hip_bf16.h>
#include <math.h>

typedef __attribute__((ext_vector_type(2))) float v2f;
typedef __attribute__((ext_vector_type(8))) float v8f;

__device__ __forceinline__ v8f wmma_f32(v2f a, v2f b, v8f c) {
  // D = A(16x4 f32) * B(4x16 f32) + C(16x16 f32)
  return __builtin_amdgcn_wmma_f32_16x16x4_f32(false, a, false, b, (short)0, c, false, false);
}

#define BB 128
#define CC 128
#define TT 64
#define VV 25
#define RR 16
#define OO 256
#define TH2 32
#define PP 800  // TH2*VV pixels per (b,c)

// ---------------- Kernel 1: sf = mean(g1, axis=T), tf = mean(g1, axis=V) ----
__global__ void k_reduce(const float* __restrict__ g1,
                         float* __restrict__ sf, float* __restrict__ tf) {
  int bc = blockIdx.x;                       // b*C + c
  __shared__ float l[TT * VV];
  const float* src = g1 + (size_t)bc * (TT * VV);
  for (int i = threadIdx.x; i < TT * VV; i += blockDim.x) l[i] = src[i];
  __syncthreads();
  int tid = threadIdx.x;
  if (tid < VV) {
    float s = 0.f;
    for (int t = 0; t < TT; ++t) s += l[t * VV + tid];
    sf[(size_t)bc * VV + tid] = s * (1.0f / (float)TT);
  }
  if (tid >= 32 && tid < 32 + TT) {
    int t = tid - 32;
    float s = 0.f;
    for (int v = 0; v < VV; ++v) s += l[t * VV + v];
    tf[(size_t)bc * TT + t] = s * (1.0f / (float)VV);
  }
}

// ---------------- Kernel 2: adjacency matrices (per batch) ------------------
__global__ void k_adj(const float* __restrict__ sf, const float* __restrict__ tf,
                      const float* __restrict__ As_r, const float* __restrict__ At_r,
                      const float* __restrict__ sc1_w, const float* __restrict__ sc1_b,
                      const float* __restrict__ sc2_w, const float* __restrict__ sc2_b,
                      const float* __restrict__ tc1_w, const float* __restrict__ tc1_b,
                      const float* __restrict__ tc2_w, const float* __restrict__ tc2_b,
                      const float* __restrict__ act_s, const float* __restrict__ act_t,
                      float* __restrict__ As_ws, float* __restrict__ At_ws,
                      float* __restrict__ o_As_at, float* __restrict__ o_At_at,
                      float* __restrict__ o_As_act, float* __restrict__ o_At_act) {
  int b = blockIdx.x;
  __shared__ float sfl[CC * VV];
  __shared__ float tfl[CC * TT];
  __shared__ float s1l[VV * RR], s2l[RR * VV];
  __shared__ float t1l[TT * RR], t2l[RR * TT];
  for (int i = threadIdx.x; i < CC * VV; i += 256) sfl[i] = sf[(size_t)b * CC * VV + i];
  for (int i = threadIdx.x; i < CC * TT; i += 256) tfl[i] = tf[(size_t)b * CC * TT + i];
  __syncthreads();
  for (int e = threadIdx.x; e < VV * RR; e += 256) {
    int v = e / RR, r = e % RR;
    float s = 0.f; const float* w = sc1_w + r * CC;
    for (int c = 0; c < CC; ++c) s += w[c] * sfl[c * VV + v];
    s1l[v * RR + r] = s + sc1_b[r];
  }
  for (int e = threadIdx.x; e < RR * VV; e += 256) {
    int r = e / VV, v = e % VV;
    float s = 0.f; const float* w = sc2_w + r * CC;
    for (int c = 0; c < CC; ++c) s += w[c] * sfl[c * VV + v];
    s2l[r * VV + v] = s + sc2_b[r];
  }
  for (int e = threadIdx.x; e < TT * RR; e += 256) {
    int t = e / RR, r = e % RR;
    float s = 0.f; const float* w = tc1_w + r * CC;
    for (int c = 0; c < CC; ++c) s += w[c] * tfl[c * TT + t];
    t1l[t * RR + r] = s + tc1_b[r];
  }
  for (int e = threadIdx.x; e < RR * TT; e += 256) {
    int r = e / TT, t = e % TT;
    float s = 0.f; const float* w = tc2_w + r * CC;
    for (int c = 0; c < CC; ++c) s += w[c] * tfl[c * TT + t];
    t2l[r * TT + t] = s + tc2_b[r];
  }
  __syncthreads();
  for (int e = threadIdx.x; e < VV * VV; e += 256) {
    int v = e / VV, u = e % VV;
    float s = 0.f;
    for (int r = 0; r < RR; ++r) s += s1l[v * RR + r] * s2l[r * VV + u];
    float at = tanhf(s); at = at > 0.f ? at : 0.f;
    float ac = As_r[(size_t)b * VV * VV + e] * act_s[e];
    o_As_at[(size_t)b * VV * VV + e] = at;
    o_As_act[(size_t)b * VV * VV + e] = ac;
    As_ws[(size_t)b * VV * VV + e] = at + ac;
  }
  for (int e = threadIdx.x; e < TT * TT; e += 256) {
    int t = e / TT, y = e % TT;
    float s = 0.f;
    for (int r = 0; r < RR; ++r) s += t1l[t * RR + r] * t2l[r * TT + y];
    float at = tanhf(s); at = at > 0.f ? at : 0.f;
    float ac = At_r[(size_t)b * TT * TT + e] * act_t[e];
    o_At_at[(size_t)b * TT * TT + e] = at;
    o_At_act[(size_t)b * TT * TT + e] = ac;
    At_ws[(size_t)b * TT * TT + e] = at + ac;
  }
}

// ---------------- Kernel 3: spatial+temporal aggregation (WMMA f32) ---------
// per (b,c): tmp[t][u] = sum_v g1[t][v]*As[u][v];  out[y][u] = sum_t tmp[t][u]*At[t][2y]
__global__ void k_stgc(const float* __restrict__ g1, const float* __restrict__ As,
                       const float* __restrict__ At, float* __restrict__ nu1a) {
  int bc = blockIdx.x;
  int b = bc >> 7;
  __shared__ float g1l[TT * 28];   // [t][v] padded K=28
  __shared__ float Asl[32 * 28];   // [u][v] padded N=32, K=28
  __shared__ float AtT[32 * 66];   // [yi][t] padded stride 66
  __shared__ float tmpT[32 * 66];  // [u][t]  padded stride 66
  const float* g1p = g1 + (size_t)bc * (TT * VV);
  for (int e = threadIdx.x; e < TT * 28; e += 256) {
    int t = e / 28, v = e % 28;
    g1l[e] = (v < VV) ? g1p[t * VV + v] : 0.0f;
  }
  const float* Asp = As + (size_t)b * VV * VV;
  for (int e = threadIdx.x; e < 32 * 28; e += 256) {
    int u = e / 28, v = e % 28;
    Asl[e] = (u < VV && v < VV) ? Asp[u * VV + v] : 0.0f;
  }
  const float* Atp = At + (size_t)b * TT * TT;
  for (int e = threadIdx.x; e < 32 * 66; e += 256) {
    int y = e / 66, t = e % 66;
    AtT[e] = (t < TT) ? Atp[t * TT + 2 * y] : 0.0f;
  }
  __syncthreads();
  int lane = threadIdx.x & 31;
  int wave = threadIdx.x >> 5;
  int lm = lane & 15, lh = lane >> 4;
  { // GEMM1: M=64(t) x N=32(u) x K=28(v); 8 waves = 4x2 tiles
    int m0 = (wave >> 1) * 16, n0 = (wave & 1) * 16;
    v8f acc = {};
    for (int k0 = 0; k0 < 28; k0 += 4) {
      v2f a  = *(const v2f*)&g1l[(m0 + lm) * 28 + k0 + 2 * lh];
      v2f bb = *(const v2f*)&Asl[(n0 + lm) * 28 + k0 + 2 * lh];
      acc = wmma_f32(a, bb, acc);
    }
#pragma unroll
    for (int i = 0; i < 8; ++i) {
      int t = m0 + i + 8 * lh, u = n0 + lm;
      tmpT[u * 66 + t] = acc[i];
    }
  }
  __syncthreads();
  if (wave < 4) { // GEMM2: M=32(y) x N=32(u) x K=64(t); 2x2 tiles
    int m0 = (wave >> 1) * 16, n0 = (wave & 1) * 16;
    v8f acc = {};
    for (int k0 = 0; k0 < TT; k0 += 4) {
      v2f a  = *(const v2f*)&AtT[(m0 + lm) * 66 + k0 + 2 * lh];
      v2f bb = *(const v2f*)&tmpT[(n0 + lm) * 66 + k0 + 2 * lh];
      acc = wmma_f32(a, bb, acc);
    }
    float* dst = nu1a + (size_t)bc * PP;
#pragma unroll
    for (int i = 0; i < 8; ++i) {
      int y = m0 + i + 8 * lh, u = n0 + lm;
      if (u < VV) dst[y * VV + u] = acc[i];
    }
  }
}

// ---------------- Kernel 4: sp_w GEMM + BN, concat h2, W1/W2 gated GEMM -----
__global__ void k_out(const float* __restrict__ nu1a, const float* __restrict__ h2,
                      const float* __restrict__ sp_w, const float* __restrict__ sp_b,
                      const float* __restrict__ sp_gamma, const float* __restrict__ sp_beta,
                      const float* __restrict__ sp_mean, const float* __restrict__ sp_var,
                      const float* __restrict__ W1_w, const float* __restrict__ W1_b,
                      const float* __restrict__ W2_w, const float* __restrict__ W2_b,
                      const float* __restrict__ bn_gamma, const float* __restrict__ bn_beta,
                      const float* __restrict__ bn_mean, const float* __restrict__ bn_var,
                      float* __restrict__ g2) {
  int tile = blockIdx.x % (PP / 16);
  int b = blockIdx.x / (PP / 16);
  int p0 = tile * 16;
  __shared__ float Xt[16][130];   // X^T  : [pixel][c]   (c = 0..127)
  __shared__ float Zt[16][258];   // Z^T  : [pixel][c2]  (c2 = 0..255)
  for (int e = threadIdx.x; e < CC * 16; e += 256) {
    int c = e >> 4, pl = e & 15;
    Xt[pl][c]       = nu1a[((size_t)b * CC + c) * PP + p0 + pl];
    Zt[pl][CC + c]  = h2[((size_t)b * CC + c) * PP + p0 + pl];
  }
  __syncthreads();
  int lane = threadIdx.x & 31;
  int wave = threadIdx.x >> 5;
  int lm = lane & 15, lh = lane >> 4;
  { // stage 1: Y1 = bn(sp_w @ X + sp_b); wave -> output rows m0..m0+15
    int m0 = wave * 16;
    v8f acc = {};
    for (int k0 = 0; k0 < CC; k0 += 4) {
      v2f a  = *(const v2f*)&sp_w[(m0 + lm) * CC + k0 + 2 * lh];
      v2f bb = *(const v2f*)&Xt[lm][k0 + 2 * lh];
      acc = wmma_f32(a, bb, acc);
    }
#pragma unroll
    for (int i = 0; i < 8; ++i) {
      int oc = m0 + i + 8 * lh;
      float scale = sp_gamma[oc] * rsqrtf(sp_var[oc] + 1e-5f);
      Zt[lm][oc] = (acc[i] + sp_b[oc] - sp_mean[oc]) * scale + sp_beta[oc];
    }
  }
  __syncthreads();
  // stage 3: a = W1@Z, g = W2@Z, out = bn(a * sigmoid(g))
  for (int ot = wave; ot < OO / 16; ot += 8) {
    int m0 = ot * 16;
    v8f acc_a = {};
    v8f acc_g = {};
    for (int k0 = 0; k0 < 2 * CC; k0 += 4) {
      v2f bb = *(const v2f*)&Zt[lm][k0 + 2 * lh];
      v2f a1 = *(const v2f*)&W1_w[(m0 + lm) * (2 * CC) + k0 + 2 * lh];
      v2f a2 = *(const v2f*)&W2_w[(m0 + lm) * (2 * CC) + k0 + 2 * lh];
      acc_a = wmma_f32(a1, bb, acc_a);
      acc_g = wmma_f32(a2, bb, acc_g);
    }
#pragma unroll
    for (int i = 0; i < 8; ++i) {
      int o = m0 + i + 8 * lh;
      float a = acc_a[i] + W1_b[o];
      float g = acc_g[i] + W2_b[o];
      float val = a * (1.0f / (1.0f + __expf(-g)));
      float scale = bn_gamma[o] * rsqrtf(bn_var[o] + 1e-5f);
      val = (val - bn_mean[o]) * scale + bn_beta[o];
      g2[((size_t)b * OO + o) * PP + p0 + lm] = val;
    }
  }
}

extern "C" void kernel_launch(void* const* d_in, const int* in_sizes, int n_in,
                              void* d_out, int out_size, void* d_ws, size_t ws_size,
                              hipStream_t stream) {
  (void)in_sizes; (void)n_in; (void)out_size; (void)ws_size;
  const float* g1    = (const float*)d_in[0];
  const float* h2    = (const float*)d_in[1];
  const float* As_r  = (const float*)d_in[2];
  const float* At_r  = (const float*)d_in[3];
  const float* sc1_w = (const float*)d_in[4];
  const float* sc1_b = (const float*)d_in[5];
  const float* sc2_w = (const float*)d_in[6];
  const float* sc2_b = (const float*)d_in[7];
  const float* tc1_w = (const float*)d_in[8];
  const float* tc1_b = (const float*)d_in[9];
  const float* tc2_w = (const float*)d_in[10];
  const float* tc2_b = (const float*)d_in[11];
  const float* act_s = (const float*)d_in[12];
  const float* act_t = (const float*)d_in[13];
  const float* sp_w  = (const float*)d_in[14];
  const float* sp_b  = (const float*)d_in[15];
  const float* sp_g  = (const float*)d_in[16];
  const float* sp_bt = (const float*)d_in[17];
  const float* sp_m  = (const float*)d_in[18];
  const float* sp_v  = (const float*)d_in[19];
  const float* W1_w  = (const float*)d_in[20];
  const float* W1_b  = (const float*)d_in[21];
  const float* W2_w  = (const float*)d_in[22];
  const float* W2_b  = (const float*)d_in[23];
  const float* bn_g  = (const float*)d_in[24];
  const float* bn_bt = (const float*)d_in[25];
  const float* bn_m  = (const float*)d_in[26];
  const float* bn_v  = (const float*)d_in[27];

  float* out = (float*)d_out;
  // d_out layout (tuple, flat): g2 | As_at | At_at | As_act | At_act
  float* o_g2     = out;
  float* o_As_at  = out + (size_t)BB * OO * TH2 * VV;                  // 26,214,400
  float* o_At_at  = o_As_at + (size_t)BB * VV * VV;                    // +80,000
  float* o_As_act = o_At_at + (size_t)BB * TT * TT;                    // +524,288
  float* o_At_act = o_As_act + (size_t)BB * VV * VV;                   // +80,000

  // workspace layout (floats)
  float* ws   = (float*)d_ws;
  float* sf   = ws;                                   // B*C*V   = 409,600
  float* tf   = sf + (size_t)BB * CC * VV;            // B*C*T   = 1,048,576
  float* Asb  = tf + (size_t)BB * CC * TT;            // B*V*V   = 80,000
  float* Atb  = Asb + (size_t)BB * VV * VV;           // B*T*T   = 524,288
  float* nu1a = Atb + (size_t)BB * TT * TT;           // B*C*32*V= 13,107,200

  k_reduce<<<BB * CC, 128, 0, stream>>>(g1, sf, tf);
  k_adj<<<BB, 256, 0, stream>>>(sf, tf, As_r, At_r,
                                sc1_w, sc1_b, sc2_w, sc2_b,
                                tc1_w, tc1_b, tc2_w, tc2_b,
                                act_s, act_t,
                                Asb, Atb, o_As_at, o_At_at, o_As_act, o_At_act);
  k_stgc<<<BB * CC, 256, 0, stream>>>(g1, Asb, Atb, nu1a);
  k_out<<<BB * (PP / 16), 256, 0, stream>>>(nu1a, h2,
                                            sp_w, sp_b, sp_g, sp_bt, sp_m, sp_v,
                                            W1_w, W1_b, W2_w, W2_b,
                                            bn_g, bn_bt, bn_m, bn_v,
                                            o_g2);
}